// UnpadSequences_49151605736123
// MI455X (gfx1250) — compile-verified
//
#include <hip/hip_runtime.h>
#include <hip/hip_bf16.h>

// Problem constants (match reference)
#define NTOK   262144
#define DMODEL 256
#define NSEQ   512
#define MAXLEN 640
#define CHUNK  512                 // rows per chunk/block
#define NCHUNK (NTOK / CHUNK)      // 512 chunks
#define TPB    256                 // 8 wave32s per block
#define NWAVE  (TPB / 32)

// ---------------------------------------------------------------------------
// Pass 1: per-chunk histogram of sequence ids -> ws.hist[chunk][NSEQ]
// ---------------------------------------------------------------------------
__global__ __launch_bounds__(TPB) void k_hist(const int* __restrict__ ids,
                                              int* __restrict__ hist) {
  __shared__ int h[NSEQ];
  const int t = threadIdx.x;
  const int b = blockIdx.x;
  h[t] = 0;
  h[t + TPB] = 0;
  __syncthreads();
  const int base = b * CHUNK;
#pragma unroll
  for (int j = 0; j < CHUNK / TPB; ++j)
    atomicAdd(&h[ids[base + j * TPB + t]], 1);
  __syncthreads();
  hist[b * NSEQ + t] = h[t];
  hist[b * NSEQ + t + TPB] = h[t + TPB];
}

// ---------------------------------------------------------------------------
// Pass 2: per-id exclusive prefix across chunks (in place) + total counts
// ---------------------------------------------------------------------------
__global__ __launch_bounds__(NSEQ) void k_scan(int* __restrict__ hist,
                                               int* __restrict__ counts) {
  const int s = threadIdx.x;  // one thread per sequence id
  int run = 0;
  for (int b = 0; b < NCHUNK; ++b) {
    const int idx = b * NSEQ + s;
    const int v = hist[idx];
    hist[idx] = run;  // exclusive prefix: rows of chunk b start here
    run += v;
  }
  counts[s] = run;
}

// ---------------------------------------------------------------------------
// Pass 3: zero-fill only the unwritten tail of each group + its mask=1.0
// (slots < count are covered exactly once by the scatter kernel)
// ---------------------------------------------------------------------------
__global__ __launch_bounds__(TPB) void k_fill(const int* __restrict__ counts,
                                              float* __restrict__ padded,
                                              float* __restrict__ mask) {
  const int s = blockIdx.x;
  int cnt = counts[s];
  cnt = cnt < 0 ? 0 : (cnt > MAXLEN ? MAXLEN : cnt);
  float4* base4 = (float4*)(padded + ((size_t)s * MAXLEN + cnt) * DMODEL);
  const int total4 = (MAXLEN - cnt) * (DMODEL / 4);
  const float4 z = make_float4(0.f, 0.f, 0.f, 0.f);
  for (int idx = threadIdx.x; idx < total4; idx += TPB) base4[idx] = z;
  for (int l = cnt + (int)threadIdx.x; l < MAXLEN; l += TPB)
    mask[(size_t)s * MAXLEN + l] = 1.0f;
}

// ---------------------------------------------------------------------------
// Pass 4: ordered ranks + coalesced scatter of 1KB rows
// ---------------------------------------------------------------------------
__global__ __launch_bounds__(TPB) void k_scatter(const float* __restrict__ batch,
                                                 const int* __restrict__ ids,
                                                 const int* __restrict__ histPrefix,
                                                 float* __restrict__ padded,
                                                 float* __restrict__ mask) {
  __shared__ __align__(16) int idsLDS[CHUNK];      // 2 KB, filled by async DMA
  __shared__ int waveCnt[NWAVE * NSEQ];            // 16 KB: hist -> prefix -> counters

  const int tid = threadIdx.x;
  const int lane = tid & 31;
  const int w = tid >> 5;
  const int b = blockIdx.x;
  const int chunkBase = b * CHUNK;

  // Phase 0: CDNA5 async DMA (global -> LDS, no VGPR data) of this chunk's ids.
  if (w == 0) {
    const unsigned ldsBase = (unsigned)(size_t)(&idsLDS[0]);
    const unsigned long long gbase =
        (unsigned long long)(const void*)(ids + chunkBase);
#pragma unroll
    for (int k = 0; k < (CHUNK * 4) / 512; ++k) {  // 4 x 512B wave transfers
      const unsigned off = (unsigned)(k * 512 + lane * 16);
      const unsigned ldsaddr = ldsBase + off;
      asm volatile("global_load_async_to_lds_b128 %0, %1, %2"
                   :: "v"(ldsaddr), "v"(off), "s"(gbase)
                   : "memory");
    }
    asm volatile("s_wait_asynccnt 0" ::: "memory");
  }
  // Zero the per-wave counter table (disjoint from the async-written region).
#pragma unroll
  for (int k = 0; k < (NWAVE * NSEQ) / TPB; ++k) waveCnt[k * TPB + tid] = 0;
  __syncthreads();

  // Rows of this chunk owned by this thread: wave w owns rows [w*64, w*64+64),
  // group i covers rows w*64 + i*32 + lane  (original order preserved).
  int myid[2];
#pragma unroll
  for (int i = 0; i < 2; ++i)
    myid[i] = *(volatile int*)&idsLDS[w * 64 + i * 32 + lane];

  // Phase 1: per-wave histogram.
#pragma unroll
  for (int i = 0; i < 2; ++i) atomicAdd(&waveCnt[w * NSEQ + myid[i]], 1);
  __syncthreads();

  // Phase 2: exclusive prefix across waves, seeded with the global chunk offset.
#pragma unroll
  for (int sI = 0; sI < NSEQ / TPB; ++sI) {
    const int s = sI * TPB + tid;
    int acc = histPrefix[b * NSEQ + s];
#pragma unroll
    for (int ww = 0; ww < NWAVE; ++ww) {
      const int tmp = waveCnt[ww * NSEQ + s];
      waveCnt[ww * NSEQ + s] = acc;
      acc += tmp;
    }
  }
  __syncthreads();

  // Phase 3: ordered ranks + cooperative row copy.
#pragma unroll 1
  for (int i = 0; i < 2; ++i) {
    const int id = myid[i];
    const int rowBase = chunkBase + w * 64 + i * 32;

    // Prefetch this group's source rows (global_prefetch_b8).
    __builtin_prefetch(batch + (size_t)(rowBase + lane) * DMODEL, 0, 1);

    // Lane-match within the 32-row group (v_readlane broadcast + compare).
    unsigned mm = 0u;
#pragma unroll
    for (int j = 0; j < 32; ++j) {
      const int idj = __builtin_amdgcn_readlane(id, j);
      if (idj == id) mm |= (1u << j);
    }
    const int leader = (int)__builtin_ctz(mm);
    const int gcount = (int)__builtin_popcount(mm);
    const int rmine = (int)__builtin_popcount(mm & ((1u << lane) - 1u));

    int base = 0;
    if (lane == leader) base = atomicAdd(&waveCnt[w * NSEQ + id], gcount);
    base = __builtin_amdgcn_ds_bpermute(leader << 2, base);  // broadcast from leader
    const int rank = base + rmine;
    const int pack = (id << 16) | rank;  // rank < 2^16

    // Whole wave copies each of the 32 rows (1KB each): 2x B128 per lane.
#pragma unroll 4
    for (int r = 0; r < 32; ++r) {
      const int pk = __builtin_amdgcn_readlane(pack, r);  // uniform (SGPR)
      const int idr = pk >> 16;
      const int rkr = pk & 0xFFFF;
      const float4* src =
          (const float4*)(batch + (size_t)(rowBase + r) * DMODEL);
      const float4 a = src[lane];
      const float4 c = src[lane + 32];
      const bool nz = (a.x != 0.f) | (a.y != 0.f) | (a.z != 0.f) |
                      (a.w != 0.f) | (c.x != 0.f) | (c.y != 0.f) |
                      (c.z != 0.f) | (c.w != 0.f);
      const unsigned bal = __builtin_amdgcn_ballot_w32(nz);
      if (rkr < MAXLEN) {  // uniform guard (matches JAX OOB-drop semantics)
        float4* dst =
            (float4*)(padded + ((size_t)idr * MAXLEN + rkr) * DMODEL);
        dst[lane] = a;
        dst[lane + 32] = c;
        if (lane == 0)
          mask[(size_t)idr * MAXLEN + rkr] = (bal == 0u) ? 1.0f : 0.0f;
      }
    }
  }
}

// ---------------------------------------------------------------------------
extern "C" void kernel_launch(void* const* d_in, const int* in_sizes, int n_in,
                              void* d_out, int out_size, void* d_ws,
                              size_t ws_size, hipStream_t stream) {
  const float* batch = (const float*)d_in[0];
  const int* ids = (const int*)d_in[1];

  float* padded = (float*)d_out;                               // [S, MAXLEN, D]
  float* mask = padded + (size_t)NSEQ * MAXLEN * DMODEL;       // [S, MAXLEN]

  int* hist = (int*)d_ws;                 // [NCHUNK, NSEQ] -> exclusive prefix
  int* counts = hist + NCHUNK * NSEQ;     // [NSEQ]

  k_hist<<<NCHUNK, TPB, 0, stream>>>(ids, hist);
  k_scan<<<1, NSEQ, 0, stream>>>(hist, counts);
  k_fill<<<NSEQ, TPB, 0, stream>>>(counts, padded, mask);
  k_scatter<<<NCHUNK, TPB, 0, stream>>>(batch, ids, hist, padded, mask);
}